// ConvBlock_2439541424481
// MI455X (gfx1250) — compile-verified
//
#include <hip/hip_runtime.h>
#include <hip/hip_bf16.h>
#include <stdint.h>

// ConvBlock(K=3,pad=1,stride=1,dil=1) unfold->max(thr)->fold collapses to an
// elementwise op: out[n,c,h,w] = sum over valid (i,j) of max(x[n,c,h,w], thr[c,3i+j]).
// Streaming kernel: one TDM DMA per block (16 rows, 7 KB) -> LDS -> compute ->
// global_store_b128. Roofline: 51.4 MB total traffic @ 23.3 TB/s ~= 2.2 us.

#define Hn 112
#define Wn 112
#define Cn 64
#define ROWS_PER_BLK 16
#define THREADS 224                          // 7 waves (wave32)
#define F4_PER_THREAD 2                      // 224*2 = 448 float4 = 16 rows
#define TILE_ELEMS (ROWS_PER_BLK * Wn)       // 1792 floats = 7168 B, contiguous

typedef __attribute__((ext_vector_type(4))) unsigned int u32x4;
typedef __attribute__((ext_vector_type(4))) int          i32x4;
typedef __attribute__((ext_vector_type(8))) int          i32x8;

#if defined(__has_builtin)
# if __has_builtin(__builtin_amdgcn_tensor_load_to_lds)
#  define HAVE_TDM 1
# endif
#endif
#ifndef HAVE_TDM
# define HAVE_TDM 0
#endif

// Validity (from the unfold/fold index ranges):
//   i=0 invalid iff h==111 ; i=2 invalid iff h==0 ; i=1 always valid
//   j=0 invalid iff w==111 ; j=2 invalid iff w==0 ; j=1 always valid
// Inclusion-exclusion on the full 9-term sum.
__device__ __forceinline__ float pixel_sum(float v, const float t[9],
                                           bool jL, bool jR, bool iT, bool iB) {
  float m0 = fmaxf(v, t[0]), m1 = fmaxf(v, t[1]), m2 = fmaxf(v, t[2]);
  float m3 = fmaxf(v, t[3]), m4 = fmaxf(v, t[4]), m5 = fmaxf(v, t[5]);
  float m6 = fmaxf(v, t[6]), m7 = fmaxf(v, t[7]), m8 = fmaxf(v, t[8]);
  float s = ((m0 + m1) + (m2 + m3)) + ((m4 + m5) + (m6 + m7)) + m8;
  if (jL) s -= (m2 + m5 + m8);              // w==0   -> column j=2 invalid
  if (jR) s -= (m0 + m3 + m6);              // w==111 -> column j=0 invalid
  if (iT) {                                 // h==0   -> row i=2 invalid
    s -= (m6 + m7 + m8);
    if (jL) s += m8;                        // corner double-subtract fixups
    if (jR) s += m6;
  }
  if (iB) {                                 // h==111 -> row i=0 invalid
    s -= (m0 + m1 + m2);
    if (jL) s += m2;
    if (jR) s += m0;
  }
  return s;
}

__global__ __launch_bounds__(THREADS) void convblock_maxfold_kernel(
    const float* __restrict__ x, const float* __restrict__ thr,
    float* __restrict__ out) {
  __shared__ float tile[TILE_ELEMS];

  const int tid   = threadIdx.x;
  const int plane = blockIdx.y;             // n*Cn + c
  const int c     = plane & (Cn - 1);       // block-uniform -> scalar thr loads
  const int hbase = blockIdx.x * ROWS_PER_BLK;
  const size_t base = (size_t)plane * (Hn * Wn) + (size_t)hbase * Wn;

#if HAVE_TDM
  // One TDM DMA per block: 1792 contiguous f32 (16 rows) global -> LDS.
  if (tid == 0) {
    const uint64_t ga  = (uint64_t)(const void*)(x + base);
    const uint32_t lds = (uint32_t)(uintptr_t)(&tile[0]);
    // D# group0: [1:0]=count=1, [63:32]=lds_addr, [120:64]=global_addr, [127:126]=type=2
    u32x4 g0 = { 1u, lds, (uint32_t)ga,
                 ((uint32_t)((ga >> 32) & 0x01FFFFFFu)) | 0x80000000u };
    // D# group1: data_size=2 (4B) in [17:16]; tensor_dim0=1792 at [79:48];
    // tile_dim0=1792 at [127:112]; tensor_dim0_stride=1792 at [207:160]; rest 0.
    i32x8 g1 = { (int)0x00020000,
                 (int)(((unsigned)TILE_ELEMS & 0xFFFFu) << 16),  // tensor_dim0[15:0]
                 0,                                              // tensor_dim0 hi / dim1 lo
                 (int)((unsigned)TILE_ELEMS << 16),              // tile_dim0
                 0,                                              // tile_dim1=tile_dim2=0
                 (int)TILE_ELEMS,                                // tensor_dim0_stride lo32
                 0, 0 };
    i32x4 z4 = { 0, 0, 0, 0 };
#if defined(__clang_major__) && (__clang_major__ >= 23)
    i32x8 z8 = { 0, 0, 0, 0, 0, 0, 0, 0 };
    __builtin_amdgcn_tensor_load_to_lds(g0, g1, z4, z4, z8, 0);
#else
    __builtin_amdgcn_tensor_load_to_lds(g0, g1, z4, z4, 0);
#endif
    __builtin_amdgcn_s_wait_tensorcnt((short)0);
  }
#else
  // Fallback: cooperative b128 copy global -> LDS.
#pragma unroll
  for (int p = 0; p < F4_PER_THREAD; ++p) {
    const int f = tid + p * THREADS;
    *reinterpret_cast<float4*>(&tile[f * 4]) =
        *reinterpret_cast<const float4*>(x + base + (size_t)f * 4);
  }
#endif

  // Thresholds for this block's channel (uniform): scalar loads overlap the DMA.
  float t[9];
#pragma unroll
  for (int k = 0; k < 9; ++k) t[k] = thr[c * 9 + k];

  __syncthreads();

#pragma unroll
  for (int p = 0; p < F4_PER_THREAD; ++p) {
    const int  f  = tid + p * THREADS;      // float4 index within tile
    const int  w4 = f % (Wn / 4);
    const int  r  = f / (Wn / 4);
    const int  h  = hbase + r;
    const bool iT = (h == 0);
    const bool iB = (h == Hn - 1);

    const float4 v = *reinterpret_cast<const float4*>(&tile[r * Wn + w4 * 4]);
    float4 o;
    o.x = pixel_sum(v.x, t, (w4 == 0), false, iT, iB);
    o.y = pixel_sum(v.y, t, false, false, iT, iB);
    o.z = pixel_sum(v.z, t, false, false, iT, iB);
    o.w = pixel_sum(v.w, t, false, (w4 == (Wn / 4 - 1)), iT, iB);

    *reinterpret_cast<float4*>(out + base + (size_t)r * Wn + w4 * 4) = o;
  }
}

extern "C" void kernel_launch(void* const* d_in, const int* in_sizes, int n_in,
                              void* d_out, int out_size, void* d_ws, size_t ws_size,
                              hipStream_t stream) {
  (void)n_in; (void)out_size; (void)d_ws; (void)ws_size;
  const float* x   = (const float*)d_in[0];   // (N,64,112,112) f32
  const float* thr = (const float*)d_in[1];   // (64,9) f32
  float* out = (float*)d_out;

  const int planes = in_sizes[0] / (Hn * Wn);     // N*C = 512
  dim3 grid(Hn / ROWS_PER_BLK, planes);           // (7, 512)
  dim3 block(THREADS);                            // 224
  convblock_maxfold_kernel<<<grid, block, 0, stream>>>(x, thr, out);
}